// neuralGrU_10264971837859
// MI455X (gfx1250) — compile-verified
//
#include <hip/hip_runtime.h>
#include <math.h>

typedef float v2f __attribute__((ext_vector_type(2)));
typedef float v8f __attribute__((ext_vector_type(8)));

namespace {
constexpr int   T_SLOTS = 24;
constexpr float K_B     = 5.0f;     // B_TEMP
constexpr float K_ETA_C = 0.95f;
constexpr float K_ETA_D = 0.95f;
constexpr float K_ALPHA = 1000.0f;
constexpr float K_BIG   = 1e30f;
constexpr float K_LOG2E = 1.4426950408889634f;

// logits are pre-shifted (<= 0); native exp handles the -1e31 sentinel -> 0.
__device__ __forceinline__ void softmax3(float l0, float l1, float l2,
                                         float& o0, float& o1, float& o2) {
  float mx  = fmaxf(l0, fmaxf(l1, l2));
  float e0  = __expf(l0 - mx);
  float e1  = __expf(l1 - mx);
  float e2  = __expf(l2 - mx);
  float inv = 1.0f / (e0 + e1 + e2);
  o0 = e0 * inv; o1 = e1 * inv; o2 = e2 * inv;
}
}  // namespace

// One wave handles 16 batch elements. Lanes 0-15 own elements; lanes 16-31
// mirror them so WMMA A/B operand staging is pure per-lane selects.
__global__ __launch_bounds__(256)
void gru_fused_kernel(const float* __restrict__ pi_g,
                      const float* __restrict__ pi_p,
                      const float* __restrict__ pi_s,
                      const float* __restrict__ pi_c,
                      const float* __restrict__ pi_d,
                      const float* __restrict__ d_t,
                      const float* __restrict__ C_t,
                      float* __restrict__ out,
                      int batch) {
  const int  lane = threadIdx.x & 31;
  const int  wave = (blockIdx.x * blockDim.x + threadIdx.x) >> 5;
  const int  ebase = wave * 16;
  if (ebase >= batch) return;                     // whole-wave guard: EXEC stays full
  int e = ebase + (lane & 15);
  if (e >= batch) e = batch - 1;                  // tail clamp (benign duplicate)
  const bool lo = (lane < 16);
  const size_t eb = (size_t)e * T_SLOTS;

  // ---------------- phase 0: pse[t] = relu(pi_s)/ETA_D (reg-per-t layout) -----
  float pse[24];
  {
    const float4* p4 = reinterpret_cast<const float4*>(pi_s + eb);
#pragma unroll
    for (int j = 0; j < 6; ++j) {
      float4 v = p4[j];
      pse[4 * j + 0] = fmaxf(v.x, 0.f) * (1.0f / K_ETA_D);
      pse[4 * j + 1] = fmaxf(v.y, 0.f) * (1.0f / K_ETA_D);
      pse[4 * j + 2] = fmaxf(v.z, 0.f) * (1.0f / K_ETA_D);
      pse[4 * j + 3] = fmaxf(v.w, 0.f) * (1.0f / K_ETA_D);
    }
  }

  // ---------------- phase 1: cs = Lstrict @ pse via f32 WMMA ------------------
  // D(tout x elem) = A(tout x tin, strict-lower ones) * B(tin x elem).
  // A 16x4 layout: lanes0-15 regs={K0,K1}, lanes16-31 regs={K2,K3}.
  // B 4x16 layout: reg0={K0 | K2}, reg1={K1 | K3} across lane halves.
  float cs[24];
  {
    v8f acc0 = {0.f, 0.f, 0.f, 0.f, 0.f, 0.f, 0.f, 0.f};
    v8f acc1 = {0.f, 0.f, 0.f, 0.f, 0.f, 0.f, 0.f, 0.f};
    const int mrow = lane & 15;
    const int ksel = (lane >= 16) ? 2 : 0;
#pragma unroll
    for (int c = 0; c < 6; ++c) {
      // Pull chunk values into SSA temporaries FIRST so the half-wave select
      // is a single v_cndmask, not a dynamically-indexed register waterfall.
      const float p0 = pse[4 * c + 0];
      const float p1 = pse[4 * c + 1];
      const float p2 = pse[4 * c + 2];
      const float p3 = pse[4 * c + 3];
      v2f b;
      b.x = lo ? p0 : p2;
      b.y = lo ? p1 : p3;
      const int kb = 4 * c + ksel;
      v2f a0, a1;
      a0.x = (kb     < mrow)      ? 1.f : 0.f;   // M-tile 0: tout = mrow
      a0.y = (kb + 1 < mrow)      ? 1.f : 0.f;
      a1.x = (kb     < mrow + 16) ? 1.f : 0.f;   // M-tile 1: tout = mrow+16
      a1.y = (kb + 1 < mrow + 16) ? 1.f : 0.f;
      acc0 = __builtin_amdgcn_wmma_f32_16x16x4_f32(false, a0, false, b,
                                                   (short)0, acc0, false, false);
      acc1 = __builtin_amdgcn_wmma_f32_16x16x4_f32(false, a1, false, b,
                                                   (short)0, acc1, false, false);
    }
    // D layout: reg v = {tout v | tout v+8} (tile0), {tout 16+v | garbage} (tile1)
#pragma unroll
    for (int v = 0; v < 8; ++v) {
      const float s0 = acc0[v];
      const float s1 = acc1[v];
      const float o0 = __shfl_xor(s0, 16, 32);
      const float o1 = __shfl_xor(s1, 16, 32);
      cs[v]      = lo ? s0 : o0;
      cs[8 + v]  = lo ? o0 : s0;
      cs[16 + v] = lo ? s1 : o1;
    }
  }
  if (!lo) return;  // mirror lanes done (WMMA staging only)

  // ---------------- phase 1b: q, exclusive prefix-min, pi_tilda --------------
  float q[24];
  {
    const float4* pc4 = reinterpret_cast<const float4*>(pi_c + eb);
    const float4* pg4 = reinterpret_cast<const float4*>(pi_g + eb);
    const float sc = 1.0f / (K_ETA_C * K_ETA_D);
#pragma unroll
    for (int j = 0; j < 6; ++j) {
      float4 c = pc4[j];
      float4 g = pg4[j];
      q[4 * j + 0] = fmaxf(c.x, 0.f) * sc + g.x - cs[4 * j + 0];
      q[4 * j + 1] = fmaxf(c.y, 0.f) * sc + g.y - cs[4 * j + 1];
      q[4 * j + 2] = fmaxf(c.z, 0.f) * sc + g.z - cs[4 * j + 2];
      q[4 * j + 3] = fmaxf(c.w, 0.f) * sc + g.w - cs[4 * j + 3];
    }
  }
  float ptld[24];
  float runmin = K_BIG;
#pragma unroll
  for (int t = 0; t < T_SLOTS; ++t) {
    ptld[t] = cs[t] + runmin;                 // pi_tilda[t] = cs[t] + min_{i<t} q[i]
    runmin = fminf(runmin, q[t]);
  }
  const float qmin = runmin;

  // w[j] = exp(-B*(q[j]-qmin)) in fp64 (wide range), W = exclusive prefix sum.
  double wD[24], WD[24];
  {
    double run = 0.0;
#pragma unroll
    for (int j = 0; j < T_SLOTS; ++j) {
      float y  = (qmin - q[j]) * (K_B * K_LOG2E);   // <= 0, base-2 exponent
      float kf = floorf(y);
      float fr = y - kf;                            // [0,1)
      double w = ldexp((double)exp2f(fr), (int)kf);
      wD[j] = w;
      WD[j] = run;
      run += w;
    }
  }

  // ---------------- phase 2: per-t cascaded softmins & demand split ----------
  float dgA[24], dd[24];
  const float* Ce = C_t + eb * 3;
#pragma unroll
  for (int t = 0; t < T_SLOTS; ++t) {
    float pp = fmaxf(pi_p[eb + t], 0.f);
    float pd = fmaxf(pi_d[eb + t], 0.f);
    float pg = pi_g[eb + t];
    float dt = d_t[eb + t];
    float C0 = Ce[3 * t + 0], C1 = Ce[3 * t + 1], C2 = Ce[3 * t + 2];

    float X0 = ptld[t] + pd * (1.0f / K_ETA_D);
    float X1 = pp;
    float X2 = pg;

    float a10, a11, a12, a20, a21, a22, a30, a31, a32;
    softmax3(-2.f * K_B * X0, -2.f * K_B * X1, -2.f * K_B * X2, a10, a11, a12);
    float Y0 = X0 + K_ALPHA * a10, Y1 = X1 + K_ALPHA * a11, Y2 = X2 + K_ALPHA * a12;
    softmax3(-2.f * K_B * Y0, -2.f * K_B * Y1, -2.f * K_B * Y2, a20, a21, a22);
    float Z0 = Y0 + K_ALPHA * a20, Z1 = Y1 + K_ALPHA * a21, Z2 = Y2 + K_ALPHA * a22;
    softmax3(-2.f * K_B * Z0, -2.f * K_B * Z1, -2.f * K_B * Z2, a30, a31, a32);

    float rem  = dt;
    float del0 = a10 * C0 + a11 * C1 + a12 * C2;
    float d0   = fminf(del0, rem); rem -= d0;     // delta - relu(delta - rem)
    float del1 = a20 * C0 + a21 * C1 + a22 * C2;
    float d1   = fminf(del1, rem); rem -= d1;
    float del2 = a30 * C0 + a31 * C1 + a32 * C2;
    float d2   = fminf(del2, rem);

    dgA[t] = a12 * d0 + a22 * d1 + a32 * d2;
    float dp = a11 * d0 + a21 * d1 + a31 * d2;
    dd[t]  = (a10 * d0 + a20 * d1 + a30 * d2) * (1.0f / K_ETA_D);

    out[(size_t)e * 120 + 1 * 24 + t] = dp;       // d_p
  }

  // ---------------- phase 3: suffix scan S[i] = sum_{t>i} dd[t]/W[t] ---------
  // d_c[i] = w[i]*S[i]/ETA_C ; d_s[i] = (W[i]+w[i])*S[i]
  double S = 0.0;
#pragma unroll
  for (int i = T_SLOTS - 1; i >= 0; --i) {
    if (i < T_SLOTS - 1) S += (double)dd[i + 1] / WD[i + 1];  // W[t>=1] > 0
    float dc = (float)(wD[i] * S * (1.0 / (double)K_ETA_C));
    float ds = (float)((WD[i] + wD[i]) * S);
    const size_t ob = (size_t)e * 120;
    out[ob + 0 * 24 + i] = dgA[i] + dc;   // d_g + d_c
    out[ob + 2 * 24 + i] = ds;            // d_s
    out[ob + 3 * 24 + i] = dc;            // d_c
    out[ob + 4 * 24 + i] = dd[i];         // d_d
  }
}

extern "C" void kernel_launch(void* const* d_in, const int* in_sizes, int n_in,
                              void* d_out, int out_size, void* d_ws, size_t ws_size,
                              hipStream_t stream) {
  (void)n_in; (void)out_size; (void)d_ws; (void)ws_size;
  const float* pi_g = (const float*)d_in[0];
  const float* pi_p = (const float*)d_in[1];
  const float* pi_s = (const float*)d_in[2];
  const float* pi_c = (const float*)d_in[3];
  const float* pi_d = (const float*)d_in[4];
  const float* d_t  = (const float*)d_in[5];
  const float* C_t  = (const float*)d_in[6];
  float* out = (float*)d_out;

  const int batch  = in_sizes[0] / 24;
  const int waves  = (batch + 15) / 16;
  const int blocks = (waves + 7) / 8;   // 8 waves (256 threads) per block
  gru_fused_kernel<<<blocks, 256, 0, stream>>>(pi_g, pi_p, pi_s, pi_c, pi_d,
                                               d_t, C_t, out, batch);
}